// QuantumPolicyNetwork_53034256171238
// MI455X (gfx1250) — compile-verified
//
#include <hip/hip_runtime.h>
#include <hip/hip_bf16.h>
#include <math.h>

// ---------------------------------------------------------------------------
// Quantum policy network on MI455X (gfx1250, wave32).
// State: 2^21 complex64 amplitudes, stored as planar float arrays (R, I).
// Qubit q <-> index bit (20 - q)  (wire 0 = MSB, PennyLane convention).
//
// Strategy:
//   * fuse Rz*Ry*Rx per (layer,qubit) -> 84 complex 2x2 unitaries (device)
//   * kron 4 consecutive qubits -> 16x16 complex unitary per (layer,group)
//   * apply 16x16 unitary with V_WMMA_F32_16X16X4_F32 (4 real GEMMs/complex)
//   * collapse the 21-CNOT entangling chain into ONE bit-permutation pass
//   * qubit 20 (index bit 0) gets a simple adjacent-pair gate kernel
// ---------------------------------------------------------------------------

#define NQ 21
#define NL 4
#define NSTATE (1u << NQ)          // 2^21
#define ACTION_DIM 18

typedef float v2f __attribute__((ext_vector_type(2)));
typedef float v8f __attribute__((ext_vector_type(8)));

struct c32 { float re, im; };
__device__ __forceinline__ c32 cmul(c32 a, c32 b) {
    return { a.re * b.re - a.im * b.im, a.re * b.im + a.im * b.re };
}
__device__ __forceinline__ c32 cadd(c32 a, c32 b) { return { a.re + b.re, a.im + b.im }; }

// ---------------------------------------------------------------------------
// 0) zero small accumulators: W[0]=norm^2 acc, W[1..21]=meas
// ---------------------------------------------------------------------------
__global__ void k_zero_small(float* W) {
    if (threadIdx.x < 64) W[threadIdx.x] = 0.0f;
}

// ---------------------------------------------------------------------------
// 1) sum of squares of input state (grid-stride + block reduce + atomic)
// ---------------------------------------------------------------------------
__global__ void k_sumsq(const float* __restrict__ x, float* __restrict__ W) {
    __shared__ float sdata[256];
    float acc = 0.0f;
    for (unsigned i = blockIdx.x * blockDim.x + threadIdx.x; i < NSTATE;
         i += gridDim.x * blockDim.x) {
        float v = x[i];
        acc += v * v;
    }
    sdata[threadIdx.x] = acc;
    __syncthreads();
    for (unsigned s = 128; s > 0; s >>= 1) {
        if (threadIdx.x < s) sdata[threadIdx.x] += sdata[threadIdx.x + s];
        __syncthreads();
    }
    if (threadIdx.x == 0) atomicAdd(W, sdata[0]);
}

// ---------------------------------------------------------------------------
// 2) amplitude encoding:  psi = state/||state||  (or |0...0> if norm == 0)
// ---------------------------------------------------------------------------
__global__ void k_encode(const float* __restrict__ x, const float* __restrict__ W,
                         float* __restrict__ R, float* __restrict__ I) {
    unsigned i = blockIdx.x * blockDim.x + threadIdx.x;
    float norm = sqrtf(W[0]);
    float inv = (norm > 0.0f) ? 1.0f / fmaxf(norm, 1e-30f) : 0.0f;
    float r = (norm > 0.0f) ? x[i] * inv : ((i == 0) ? 1.0f : 0.0f);
    R[i] = r;
    I[i] = 0.0f;
}

// ---------------------------------------------------------------------------
// 3) fused per-(layer,qubit) unitary  U = Rz(tz) * Ry(ty) * Rx(tx)
//    fU layout: [(layer*21+q)*8 + (i*2+j)*2 + {re,im}]
// ---------------------------------------------------------------------------
__global__ void k_build_fused(const float* __restrict__ p, float* __restrict__ fU) {
    int t = blockIdx.x * blockDim.x + threadIdx.x;
    if (t >= NL * NQ) return;
    float tx = p[t * 3 + 0], ty = p[t * 3 + 1], tz = p[t * 3 + 2];
    float cx = cosf(tx * 0.5f), sx = sinf(tx * 0.5f);
    float cy = cosf(ty * 0.5f), sy = sinf(ty * 0.5f);
    float cz = cosf(tz * 0.5f), sz = sinf(tz * 0.5f);

    c32 RX[2][2] = { { {cx, 0}, {0, -sx} }, { {0, -sx}, {cx, 0} } };
    c32 RY[2][2] = { { {cy, 0}, {-sy, 0} }, { {sy, 0}, {cy, 0} } };
    c32 RZ[2][2] = { { {cz, -sz}, {0, 0} }, { {0, 0}, {cz, sz} } };

    c32 M[2][2], U[2][2];
    for (int i = 0; i < 2; ++i)
        for (int j = 0; j < 2; ++j)
            M[i][j] = cadd(cmul(RY[i][0], RX[0][j]), cmul(RY[i][1], RX[1][j]));
    for (int i = 0; i < 2; ++i)
        for (int j = 0; j < 2; ++j)
            U[i][j] = cadd(cmul(RZ[i][0], M[0][j]), cmul(RZ[i][1], M[1][j]));

    float* o = fU + t * 8;
    for (int i = 0; i < 2; ++i)
        for (int j = 0; j < 2; ++j) {
            o[(i * 2 + j) * 2 + 0] = U[i][j].re;
            o[(i * 2 + j) * 2 + 1] = U[i][j].im;
        }
}

// ---------------------------------------------------------------------------
// 4) group unitary G = U_{4g} (x) U_{4g+1} (x) U_{4g+2} (x) U_{4g+3}  (16x16)
//    gU layout per (layer,group): Gr[256] then Gi[256]   (stride 512)
// ---------------------------------------------------------------------------
__global__ void k_build_group(const float* __restrict__ fU, float* __restrict__ gU) {
    int t = blockIdx.x * blockDim.x + threadIdx.x;   // 4*5*256 = 5120 threads
    if (t >= NL * 5 * 256) return;
    int layer = t / 1280;
    int rem   = t % 1280;
    int g     = rem / 256;
    int e     = rem % 256;
    int i = e >> 4, j = e & 15;

    c32 acc = {1.0f, 0.0f};
    #pragma unroll
    for (int k = 0; k < 4; ++k) {
        int q  = 4 * g + k;
        int ib = (i >> (3 - k)) & 1;   // qubit 4g is the top bit of the field
        int jb = (j >> (3 - k)) & 1;
        const float* u = fU + (layer * NQ + q) * 8 + (ib * 2 + jb) * 2;
        acc = cmul(acc, c32{u[0], u[1]});
    }
    float* o = gU + (layer * 5 + g) * 512;
    o[i * 16 + j]       = acc.re;
    o[256 + i * 16 + j] = acc.im;
}

// ---------------------------------------------------------------------------
// 5) WMMA 16x16 group-gate, in-place on (R, I).
//    Field occupies index bits [p, p+3].  State viewed as (h, j, l):
//        idx = (h << (p+4)) + (j << p) + l,   out[i] = sum_j G[i][j] in[j]
//    Complex via 4 real GEMMs: Dr = Ur*Br - Ui*Bi ; Di = Ur*Bi + Ui*Br,
//    realized with 16x V_WMMA_F32_16X16X4_F32 per wave (K=16 split into 4).
//    Launch: grid 1024 x 256 threads = 8192 waves = 2^17/16 column tiles.
// ---------------------------------------------------------------------------
__global__ void k_wmma_gate(float* __restrict__ R, float* __restrict__ I,
                            const float* __restrict__ G, int p) {
    const float* Gr = G;
    const float* Gi = G + 256;

    const int lane   = threadIdx.x & 31;
    const int wave   = threadIdx.x >> 5;
    const int tile   = blockIdx.x * (blockDim.x >> 5) + wave;
    const unsigned n0 = (unsigned)tile << 4;

    const int M     = lane & 15;         // A row (all lanes)
    const int khalf = (lane >> 4) << 1;  // K sub-offset: 0 (lanes 0-15) / 2 (16-31)

    // --- A fragments: Ur, Ui, -Ui  (f32 WMMA has no A-negate; NEG is C-only)
    v2f aR[4], aI[4], aIn[4];
    #pragma unroll
    for (int kc = 0; kc < 4; ++kc) {
        int kb = 4 * kc + khalf;
        aR[kc].x = Gr[M * 16 + kb];     aR[kc].y = Gr[M * 16 + kb + 1];
        aI[kc].x = Gi[M * 16 + kb];     aI[kc].y = Gi[M * 16 + kb + 1];
        aIn[kc]  = -aI[kc];
    }

    // --- this lane's state column
    const unsigned col  = (unsigned)(lane & 15);
    const unsigned n    = n0 + col;
    const unsigned mask = (1u << p) - 1u;
    const unsigned h    = n >> p;
    const unsigned l    = n & mask;
    const unsigned base = (h << (p + 4)) + l;

    __builtin_prefetch(R + base, 0, 3);   // global_prefetch_b8
    __builtin_prefetch(I + base, 0, 3);

    // --- B fragments: rows j = 4*kc + khalf, 4*kc + khalf + 1 at this column
    v2f bR[4], bI[4];
    #pragma unroll
    for (int kc = 0; kc < 4; ++kc) {
        unsigned j  = 4u * kc + (unsigned)khalf;
        unsigned a0 = base + (j << p);
        unsigned a1 = base + ((j + 1u) << p);
        bR[kc].x = R[a0];  bR[kc].y = R[a1];
        bI[kc].x = I[a0];  bI[kc].y = I[a1];
    }

    // --- complex GEMM accumulation (16 WMMAs)
    v8f dr = {};
    v8f di = {};
    #pragma unroll
    for (int kc = 0; kc < 4; ++kc) {
        dr = __builtin_amdgcn_wmma_f32_16x16x4_f32(false, aR[kc],  false, bR[kc],
                                                   (short)0, dr, false, false);
        dr = __builtin_amdgcn_wmma_f32_16x16x4_f32(false, aIn[kc], false, bI[kc],
                                                   (short)0, dr, false, false);
        di = __builtin_amdgcn_wmma_f32_16x16x4_f32(false, aR[kc],  false, bI[kc],
                                                   (short)0, di, false, false);
        di = __builtin_amdgcn_wmma_f32_16x16x4_f32(false, aI[kc],  false, bR[kc],
                                                   (short)0, di, false, false);
    }

    // --- store D (lane < 16: M = r, col = lane; lane >= 16: M = r+8, col = lane-16)
    #pragma unroll
    for (int r = 0; r < 8; ++r) {
        unsigned Mo   = (unsigned)r + ((lane >> 4) << 3);
        unsigned addr = (h << (p + 4)) + (Mo << p) + l;
        R[addr] = dr[r];
        I[addr] = di[r];
    }
}

// ---------------------------------------------------------------------------
// 6) qubit-20 gate (index bit 0): adjacent pairs, in-place. u -> 8 floats.
// ---------------------------------------------------------------------------
__global__ void k_gate_q20(float* __restrict__ R, float* __restrict__ I,
                           const float* __restrict__ u) {
    unsigned t  = blockIdx.x * blockDim.x + threadIdx.x;  // 2^20 threads
    unsigned i0 = t * 2u, i1 = i0 + 1u;
    c32 u00{u[0], u[1]}, u01{u[2], u[3]}, u10{u[4], u[5]}, u11{u[6], u[7]};
    c32 z0{R[i0], I[i0]}, z1{R[i1], I[i1]};
    c32 o0 = cadd(cmul(u00, z0), cmul(u01, z1));
    c32 o1 = cadd(cmul(u10, z0), cmul(u11, z1));
    R[i0] = o0.re; I[i0] = o0.im;
    R[i1] = o1.re; I[i1] = o1.im;
}

// ---------------------------------------------------------------------------
// 7) entangling layer: CNOT(0,1)..CNOT(19,20), CNOT(20,0) collapsed into one
//    basis permutation.  Gather form: out[y] = in[x(y)] with
//      x bits 0..18  = (y ^ (y>>1)) bits 0..18
//      x bit 20      = y20 ^ y0
//      x bit 19      = y19 ^ y20 ^ y0
// ---------------------------------------------------------------------------
__global__ void k_cnot_chain(const float* __restrict__ Rin, const float* __restrict__ Iin,
                             float* __restrict__ Rout, float* __restrict__ Iout) {
    unsigned y = blockIdx.x * blockDim.x + threadIdx.x;
    unsigned low   = (y ^ (y >> 1)) & 0x7FFFFu;
    unsigned y0    = y & 1u;
    unsigned y19   = (y >> 19) & 1u;
    unsigned y20   = (y >> 20) & 1u;
    unsigned b20   = y20 ^ y0;
    unsigned b19   = y19 ^ y20 ^ y0;
    unsigned x     = low | (b20 << 20) | (b19 << 19);
    Rout[y] = Rin[x];
    Iout[y] = Iin[x];
}

// ---------------------------------------------------------------------------
// 8) per-qubit <Z>: meas[q] = sum_i |psi_i|^2 * (1 - 2*bit_{20-q}(i))
//    register accumulators + LDS + global atomics, into W[1..21]
// ---------------------------------------------------------------------------
__global__ void k_measure(const float* __restrict__ R, const float* __restrict__ I,
                          float* __restrict__ W) {
    __shared__ float sm[NQ];
    if (threadIdx.x < NQ) sm[threadIdx.x] = 0.0f;
    __syncthreads();

    float acc[NQ];
    #pragma unroll
    for (int q = 0; q < NQ; ++q) acc[q] = 0.0f;

    for (unsigned i = blockIdx.x * blockDim.x + threadIdx.x; i < NSTATE;
         i += gridDim.x * blockDim.x) {
        float pr = R[i] * R[i] + I[i] * I[i];
        #pragma unroll
        for (int q = 0; q < NQ; ++q) {
            unsigned bit = (i >> (20 - q)) & 1u;
            acc[q] += bit ? -pr : pr;
        }
    }
    #pragma unroll
    for (int q = 0; q < NQ; ++q) atomicAdd(&sm[q], acc[q]);
    __syncthreads();
    if (threadIdx.x < NQ) atomicAdd(&W[1 + threadIdx.x], sm[threadIdx.x]);
}

// ---------------------------------------------------------------------------
// 9) head + softmax, single wave
// ---------------------------------------------------------------------------
__global__ void k_head(const float* __restrict__ W, const float* __restrict__ hw,
                       const float* __restrict__ hb, float* __restrict__ out) {
    __shared__ float lg[ACTION_DIM];
    const float* meas = W + 1;
    int a = threadIdx.x;
    if (a < ACTION_DIM) {
        float s = hb[a];
        #pragma unroll
        for (int q = 0; q < NQ; ++q) s += hw[a * NQ + q] * meas[q];
        lg[a] = s;
    }
    __syncthreads();
    if (a == 0) {
        float m = lg[0];
        for (int k = 1; k < ACTION_DIM; ++k) m = fmaxf(m, lg[k]);
        float sum = 0.0f;
        float e[ACTION_DIM];
        for (int k = 0; k < ACTION_DIM; ++k) { e[k] = expf(lg[k] - m); sum += e[k]; }
        float inv = 1.0f / sum;
        for (int k = 0; k < ACTION_DIM; ++k) out[k] = e[k] * inv;
    }
}

// ---------------------------------------------------------------------------
// launch
// ---------------------------------------------------------------------------
extern "C" void kernel_launch(void* const* d_in, const int* in_sizes, int n_in,
                              void* d_out, int out_size, void* d_ws, size_t ws_size,
                              hipStream_t stream) {
    const float* state  = (const float*)d_in[0];
    const float* params = (const float*)d_in[1];
    const float* head_w = (const float*)d_in[2];
    const float* head_b = (const float*)d_in[3];
    float* out = (float*)d_out;

    const unsigned N = NSTATE;
    float* W  = (float*)d_ws;         // [0]=norm acc, [1..21]=meas
    float* R0 = W + 64;
    float* I0 = R0 + N;
    float* R1 = I0 + N;
    float* I1 = R1 + N;
    float* fU = W + 64 + 4ull * N;    // 84 fused 2x2 unitaries (672 floats)
    float* gU = fU + 672;             // 20 group 16x16 unitaries (10240 floats)

    k_zero_small<<<1, 64, 0, stream>>>(W);
    k_sumsq<<<512, 256, 0, stream>>>(state, W);
    k_encode<<<N / 256, 256, 0, stream>>>(state, W, R0, I0);
    k_build_fused<<<1, 128, 0, stream>>>(params, fU);
    k_build_group<<<20, 256, 0, stream>>>(fU, gU);

    float *cR = R0, *cI = I0, *oR = R1, *oI = I1;
    for (int layer = 0; layer < NL; ++layer) {
        for (int g = 0; g < 5; ++g) {
            int p = 17 - 4 * g;  // LSB position of the 4-bit field (qubits 4g..4g+3)
            k_wmma_gate<<<1024, 256, 0, stream>>>(cR, cI, gU + (layer * 5 + g) * 512, p);
        }
        k_gate_q20<<<(N / 2) / 256, 256, 0, stream>>>(cR, cI, fU + (layer * NQ + 20) * 8);
        if (layer < NL - 1) {
            k_cnot_chain<<<N / 256, 256, 0, stream>>>(cR, cI, oR, oI);
            float* t;
            t = cR; cR = oR; oR = t;
            t = cI; cI = oI; oI = t;
        }
    }

    k_measure<<<512, 256, 0, stream>>>(cR, cI, W);
    k_head<<<1, 32, 0, stream>>>(W, head_w, head_b, out);
}